// SurvivalCLMBRTask_82583631167787
// MI455X (gfx1250) — compile-verified
//
#include <hip/hip_runtime.h>
#include <math.h>

#define M_ROWS  8192
#define BATCH   1024
#define TBINS   8
#define DIM     512
#define C_CODES 16384
#define FEAT    768
#define NOUT    4088   /* T*(DIM-1) */
#define NNZ     262144
#define NEV     65536

typedef float v2f __attribute__((ext_vector_type(2)));
typedef float v8f __attribute__((ext_vector_type(8)));

__device__ __forceinline__ v8f wmma_f32(v2f a, v2f b, v8f c) {
    // D = A(16x4) * B(4x16) + C, full fp32
    return __builtin_amdgcn_wmma_f32_16x16x4_f32(false, a, false, b,
                                                 (short)0, c, false, false);
}

// Async 16-byte copy global -> LDS (ASYNCcnt-tracked, CDNA5).
__device__ __forceinline__ void async_copy16(const float* g, float* lds_generic) {
    const unsigned lds_off = (unsigned)(unsigned long long)lds_generic;
    const unsigned long long ga = (unsigned long long)g;
    asm volatile("global_load_async_to_lds_b128 %0, %1, off"
                 :: "v"(lds_off), "v"(ga) : "memory");
}

// ---------------------------------------------------------------------------
// Zero scratch accumulators (must run every launch: graph replay)
// ---------------------------------------------------------------------------
__global__ __launch_bounds__(256) void zero_kernel(float* rowsum, float* accum) {
    int i = blockIdx.x * 256 + threadIdx.x;
    if (i < M_ROWS) rowsum[i] = 0.f;
    if (blockIdx.x == 0 && threadIdx.x < 2) accum[threadIdx.x] = 0.f;
}

// ---------------------------------------------------------------------------
// GEMM1: binned = features @ final_w + final_b, scattered into full_a layout.
// Each wave computes a 16x64 strip (4 WMMA subtiles sharing the A fragment).
// M=1024, N padded 4088->4096, K=768. Branchless inner loop (column clamp +
// 0/1 mask precomputed outside the K loop).
// ---------------------------------------------------------------------------
__global__ __launch_bounds__(256)
void gemm1_kernel(const float* __restrict__ features,
                  const float* __restrict__ final_w,
                  const float* __restrict__ final_b,
                  float* __restrict__ full_a) {
    const int wave = threadIdx.x >> 5;
    const int lane = threadIdx.x & 31;
    const int half = lane >> 4;
    const int l16  = lane & 15;
    const int strip = blockIdx.x * 8 + wave;       // 4096 strips total
    const int m0 = (strip >> 6) << 4;              // 64 M-tiles of 16
    const int n0 = (strip & 63) << 6;              // 64 N-strips of 64

    int   ncol[4];
    float msk[4];
#pragma unroll
    for (int j = 0; j < 4; ++j) {
        const int n = n0 + j * 16 + l16;
        ncol[j] = (n < NOUT) ? n : 0;
        msk[j]  = (n < NOUT) ? 1.f : 0.f;
    }

    v8f acc[4] = {};
    const float* arow = features + (size_t)(m0 + l16) * FEAT;

    for (int k = 0; k < FEAT; k += 4) {
        const int ka = k + 2 * half;               // K pair selected by half
        v2f a;
        a.x = arow[ka];
        a.y = arow[ka + 1];
#pragma unroll
        for (int j = 0; j < 4; ++j) {
            v2f b;
            b.x = final_w[(size_t)ka * NOUT + ncol[j]] * msk[j];
            b.y = final_w[(size_t)(ka + 1) * NOUT + ncol[j]] * msk[j];
            acc[j] = wmma_f32(a, b, acc[j]);
        }
    }

    // Scatter: out col n -> (t = n/511, d = n%511); full_a row = b*8 + t.
#pragma unroll
    for (int j = 0; j < 4; ++j) {
        const int n = n0 + j * 16 + l16;
        if (n < NOUT) {
            const float bias = final_b[n];
            const int t = n / 511;
            const int d = n - t * 511;
#pragma unroll
            for (int g = 0; g < 8; ++g) {
                const int mb = m0 + g + 8 * half;          // batch row
                full_a[(size_t)(mb * TBINS + t) * DIM + d] = acc[j][g] + bias;
            }
        }
    }
}

// Append the ones column (d = 511) of full_a.
__global__ __launch_bounds__(256) void ones_kernel(float* __restrict__ full_a) {
    int r = blockIdx.x * 256 + threadIdx.x;        // 8192 rows
    full_a[(size_t)r * DIM + (DIM - 1)] = 1.0f;
}

// ---------------------------------------------------------------------------
// GEMM2 + exp2 + rowsum: scores = exp2(full_a @ cw^T).
// 128x128 tile per workgroup, double-buffered LDS staged via
// GLOBAL_LOAD_ASYNC_TO_LDS_B128 (ASYNCcnt), 8 waves, each wave owns a
// 16x128 strip = 8 WMMA accumulators. Scores stored non-temporally so the
// 537 MB output stream does not evict full_a/cw from L2.
// ---------------------------------------------------------------------------
__global__ __launch_bounds__(256)
void gemm2_kernel(const float* __restrict__ full_a,
                  const float* __restrict__ cw,
                  float* __restrict__ scores,
                  float* __restrict__ rowsum) {
    __shared__ float As[2][128 * 36];              // stride 36: conflict-free
    __shared__ float Bs[2][128 * 36];

    const int tid  = threadIdx.x;
    const int wave = tid >> 5;
    const int lane = tid & 31;
    const int half = lane >> 4;
    const int l16  = lane & 15;
    const int tile_m = blockIdx.x >> 7;            // 64 M tiles
    const int tile_n = blockIdx.x & 127;           // 128 N tiles
    const int m0 = tile_m * 128;
    const int n0 = tile_n * 128;

    // Per-thread staging coordinates (4 x 16B per array per chunk).
    int srow[4], sc4[4];
#pragma unroll
    for (int i = 0; i < 4; ++i) {
        const int idx = tid + i * 256;             // 0..1023
        srow[i] = idx >> 3;
        sc4[i]  = (idx & 7) << 2;
    }

    v8f acc[8] = {};

    // Prologue: stage chunk 0 into buffer 0.
#pragma unroll
    for (int i = 0; i < 4; ++i) {
        async_copy16(full_a + (size_t)(m0 + srow[i]) * DIM + sc4[i],
                     &As[0][srow[i] * 36 + sc4[i]]);
        async_copy16(cw + (size_t)(n0 + srow[i]) * DIM + sc4[i],
                     &Bs[0][srow[i] * 36 + sc4[i]]);
    }

    int buf = 0;
    for (int kc = 0; kc < DIM; kc += 32) {
        if (kc + 32 < DIM) {
            // Stage next chunk into the other buffer, then wait for current
            // chunk (async ops complete in order: <=8 outstanding means the
            // first 8 -- the current buffer's -- are done).
#pragma unroll
            for (int i = 0; i < 4; ++i) {
                async_copy16(full_a + (size_t)(m0 + srow[i]) * DIM + (kc + 32) + sc4[i],
                             &As[buf ^ 1][srow[i] * 36 + sc4[i]]);
                async_copy16(cw + (size_t)(n0 + srow[i]) * DIM + (kc + 32) + sc4[i],
                             &Bs[buf ^ 1][srow[i] * 36 + sc4[i]]);
            }
            asm volatile("s_wait_asynccnt 0x8" ::: "memory");
        } else {
            asm volatile("s_wait_asynccnt 0x0" ::: "memory");
        }
        __syncthreads();

        const float* curA = &As[buf][0];
        const float* curB = &Bs[buf][0];
#pragma unroll
        for (int kk = 0; kk < 8; ++kk) {
            const v2f a = *reinterpret_cast<const v2f*>(
                &curA[(wave * 16 + l16) * 36 + kk * 4 + 2 * half]);
#pragma unroll
            for (int j = 0; j < 8; ++j) {
                const v2f b = *reinterpret_cast<const v2f*>(
                    &curB[(j * 16 + l16) * 36 + kk * 4 + 2 * half]);
                acc[j] = wmma_f32(a, b, acc[j]);
            }
        }
        __syncthreads();
        buf ^= 1;
    }

    // Epilogue: exp2 (v_exp_f32), single non-temporal store, fused row sums.
    float rs[8];
#pragma unroll
    for (int g = 0; g < 8; ++g) rs[g] = 0.f;
#pragma unroll
    for (int j = 0; j < 8; ++j) {
        const int n = n0 + j * 16 + l16;
#pragma unroll
        for (int g = 0; g < 8; ++g) {
            const int m = m0 + wave * 16 + g + 8 * half;
            const float v = __builtin_amdgcn_exp2f(acc[j][g]);
            __builtin_nontemporal_store(v, &scores[(size_t)m * C_CODES + n]);
            rs[g] += v;
        }
    }
    // Reduce over the 16 lanes of each half (N dimension), then atomic per row.
#pragma unroll
    for (int g = 0; g < 8; ++g) {
        float s = rs[g];
        s += __shfl_xor(s, 1, 32);
        s += __shfl_xor(s, 2, 32);
        s += __shfl_xor(s, 4, 32);
        s += __shfl_xor(s, 8, 32);
        if (l16 == 0)
            atomicAdd(&rowsum[m0 + wave * 16 + g + 8 * half], s);
    }
}

// ---------------------------------------------------------------------------
// Survival base term: sum_m defaults[m] * rowsum[m]  -> accum[1]
// ---------------------------------------------------------------------------
__global__ __launch_bounds__(256)
void base_kernel(const float* __restrict__ defaults,
                 const float* __restrict__ rowsum,
                 float* __restrict__ accum) {
    const int m = blockIdx.x * 256 + threadIdx.x;  // exactly 8192
    float s = defaults[m] * rowsum[m];
    s += __shfl_xor(s, 16, 32);
    s += __shfl_xor(s, 8, 32);
    s += __shfl_xor(s, 4, 32);
    s += __shfl_xor(s, 2, 32);
    s += __shfl_xor(s, 1, 32);
    if ((threadIdx.x & 31) == 0) atomicAdd(accum + 1, s);
}

// ---------------------------------------------------------------------------
// Sparse overrides: corr = (val - default[row]) * scores[row, col] -> accum[1]
// Row via binary search of CSR offsets (searchsorted right - 1).
// ---------------------------------------------------------------------------
__global__ __launch_bounds__(256)
void sparse_kernel(const long long* __restrict__ offs,
                   const float* __restrict__ defaults,
                   const long long* __restrict__ sidx,
                   const float* __restrict__ svals,
                   const float* __restrict__ scores,
                   float* __restrict__ accum) {
    const int i = blockIdx.x * 256 + threadIdx.x;  // exactly NNZ
    int lo = 0, hi = M_ROWS - 1;
    while (lo < hi) {
        const int mid = (lo + hi + 1) >> 1;
        if (offs[mid] <= (long long)i) lo = mid; else hi = mid - 1;
    }
    const long long col = sidx[i];
    float corr = (svals[i] - defaults[lo]) * scores[(size_t)lo * C_CODES + col];
    corr += __shfl_xor(corr, 16, 32);
    corr += __shfl_xor(corr, 8, 32);
    corr += __shfl_xor(corr, 4, 32);
    corr += __shfl_xor(corr, 2, 32);
    corr += __shfl_xor(corr, 1, 32);
    if ((threadIdx.x & 31) == 0) atomicAdd(accum + 1, corr);
}

// ---------------------------------------------------------------------------
// Event dots: one wave per event; 512-dim dot via float4 loads -> accum[0]
// ---------------------------------------------------------------------------
__global__ __launch_bounds__(256)
void event_kernel(const float* __restrict__ full_a,
                  const float* __restrict__ cw,
                  const long long* __restrict__ ev,
                  float* __restrict__ accum) {
    const int gid  = blockIdx.x * 256 + threadIdx.x;
    const int wv   = gid >> 5;                     // exactly NEV waves
    const int lane = gid & 31;
    const long long r = ev[2 * (long long)wv];
    const long long c = ev[2 * (long long)wv + 1];
    const float4* ar = reinterpret_cast<const float4*>(full_a + r * DIM);
    const float4* wr = reinterpret_cast<const float4*>(cw + c * DIM);
    float s = 0.f;
#pragma unroll
    for (int ch = 0; ch < 4; ++ch) {
        const float4 x = ar[ch * 32 + lane];
        const float4 y = wr[ch * 32 + lane];
        s += x.x * y.x + x.y * y.y + x.z * y.z + x.w * y.w;
    }
    s += __shfl_xor(s, 16, 32);
    s += __shfl_xor(s, 8, 32);
    s += __shfl_xor(s, 4, 32);
    s += __shfl_xor(s, 2, 32);
    s += __shfl_xor(s, 1, 32);
    if (lane == 0) atomicAdd(accum + 0, s);
}

// ---------------------------------------------------------------------------
// Finalize: out[0] = event_loss + survival_loss
// ---------------------------------------------------------------------------
__global__ void finalize_kernel(const unsigned char* __restrict__ mask,
                                const float* __restrict__ accum,
                                float* __restrict__ out) {
    float nm = 0.f;
    for (int i = 0; i < BATCH; ++i) nm += mask[i] ? 1.f : 0.f;
    const float ev_sum   = accum[0];
    const float surv_sum = accum[1];
    const float event_loss = -(0.69314718055994530942f * ev_sum) / (nm * (float)C_CODES);
    const float survival_loss = surv_sum / ((float)C_CODES * nm);
    out[0] = event_loss + survival_loss;
}

// ---------------------------------------------------------------------------
extern "C" void kernel_launch(void* const* d_in, const int* in_sizes, int n_in,
                              void* d_out, int out_size, void* d_ws, size_t ws_size,
                              hipStream_t stream) {
    const float*         features = (const float*)d_in[0];
    const unsigned char* mask     = (const unsigned char*)d_in[1];
    const float*         final_w  = (const float*)d_in[2];
    const float*         final_b  = (const float*)d_in[3];
    const float*         cw       = (const float*)d_in[4];
    const long long*     soff     = (const long long*)d_in[5];
    const float*         sdef     = (const float*)d_in[6];
    const long long*     sidx     = (const long long*)d_in[7];
    const float*         sval     = (const float*)d_in[8];
    const long long*     ev       = (const long long*)d_in[9];

    float* outf   = (float*)d_out;
    float* scores = outf + 1;                      // tuple: [loss, scores...]

    float* full_a = (float*)d_ws;                  // 8192*512 floats (16 MB)
    float* rowsum = full_a + (size_t)M_ROWS * DIM; // 8192 floats
    float* accum  = rowsum + M_ROWS;               // [0]=event sum, [1]=survival sum

    zero_kernel    <<<32,   256, 0, stream>>>(rowsum, accum);
    gemm1_kernel   <<<512,  256, 0, stream>>>(features, final_w, final_b, full_a);
    ones_kernel    <<<32,   256, 0, stream>>>(full_a);
    gemm2_kernel   <<<8192, 256, 0, stream>>>(full_a, cw, scores, rowsum);
    base_kernel    <<<32,   256, 0, stream>>>(sdef, rowsum, accum);
    sparse_kernel  <<<1024, 256, 0, stream>>>(soff, sdef, sidx, sval, scores, accum);
    event_kernel   <<<8192, 256, 0, stream>>>(full_a, cw, ev, accum);
    finalize_kernel<<<1,    1,   0, stream>>>(mask, accum, outf);
}